// GNNLayer_7241314861531
// MI455X (gfx1250) — compile-verified
//
#include <hip/hip_runtime.h>
#include <hip/hip_bf16.h>

// Problem constants (from the reference)
#define BATCH   8
#define NNODE   4096            // N
#define NT      (BATCH * NNODE) // 32768 nodes
#define DD      256             // feature dim D
#define K1      17
#define KEDGE   16              // K1-1 edges kept per node
#define NEDGE   (NT * KEDGE)    // 524288
#define NELEM   ((size_t)NT * DD)
#define EPS_LN  1e-5f

typedef float v2f  __attribute__((ext_vector_type(2)));
typedef float v8f  __attribute__((ext_vector_type(8)));
typedef unsigned int u32x4 __attribute__((ext_vector_type(4)));
typedef int   i32x8 __attribute__((ext_vector_type(8)));
typedef int   i32x4 __attribute__((ext_vector_type(4)));

#define KPANEL  32
#define NPANEL  (DD / KPANEL)          // 8
// LDS layout (floats) in a single __shared__ array => byte offset 0 in group seg
#define WBUF_F(i)   ((i) * KPANEL * DD)                 // 2 x 8192 floats
#define ABUF_F(i)   (2 * KPANEL * DD + (i) * 32 * KPANEL) // 2 x 1024 floats
#define SMEM_FLOATS (2 * KPANEL * DD + 2 * 32 * KPANEL)   // 18432 floats = 72KB

// Compiler fence: tell the optimizer that LDS behind `p` may have been written
// (by the Tensor Data Mover, which it cannot see). Emits no instructions.
__device__ __forceinline__ void lds_publish(void* p) {
    asm volatile("" : : "v"(p) : "memory");
}

// ---------------------------------------------------------------------------
// Tensor Data Mover: DMA a 2-D fp32 tile (tile_x cols x tile_y rows, global
// row stride = stride_elems) into LDS at byte offset lds_byte_off.
// Descriptor per CDNA5 ISA. Groups 2/3 = 0 (2D).
// This toolchain exposes the 6-arg builtin: (g0, g1, g2, g3, g4, cpol).
// ---------------------------------------------------------------------------
__device__ __forceinline__ void tdm_load_2d(const float* gsrc,
                                            unsigned lds_byte_off,
                                            int tile_x, int tile_y,
                                            int stride_elems) {
#if __has_builtin(__builtin_amdgcn_tensor_load_to_lds)
    unsigned long long ga = (unsigned long long)gsrc;
    u32x4 g0;
    g0[0] = 1u;                                           // count=1, user D#
    g0[1] = lds_byte_off;                                 // lds_addr
    g0[2] = (unsigned)(ga & 0xFFFFFFFFu);                 // global_addr[31:0]
    g0[3] = (unsigned)((ga >> 32) & 0x1FFFFFFu)           // global_addr[56:32]
          | (2u << 30);                                   // type = 2 ("image")
    i32x8 g1;
    g1[0] = (2 << 16);                                    // data_size=4B, mask=0
    g1[1] = (tile_x & 0xFFFF) << 16;                      // tensor_dim0[15:0]
    g1[2] = (int)(((unsigned)tile_x >> 16)                // tensor_dim0[31:16]
          | (((unsigned)tile_y & 0xFFFFu) << 16));        // tensor_dim1[15:0]
    g1[3] = (int)(((unsigned)tile_y >> 16)                // tensor_dim1[31:16]
          | (((unsigned)tile_x & 0xFFFFu) << 16));        // tile_dim0
    g1[4] = (tile_y & 0xFFFF);                            // tile_dim1 (dim2=0)
    g1[5] = stride_elems;                                 // tensor_dim0_stride lo
    g1[6] = 0;                                            // stride hi / dim1 stride
    g1[7] = 0;
    i32x4 z4 = {0, 0, 0, 0};
    i32x8 z8 = {0, 0, 0, 0, 0, 0, 0, 0};
    __builtin_amdgcn_tensor_load_to_lds(g0, g1, z4, z4, z8, 0);
#endif
}

// ---------------------------------------------------------------------------
// WMMA fp32 GEMM:  Out[NT x 256] = A[NT x 256 (row-stride lda)] * W[256 x 256]
// Block: 256 threads (8 waves). Block tile: 32 rows x 256 cols.
// Wave tile: 16 rows x 64 cols = 4 accumulators of V_WMMA_F32_16X16X4_F32.
// Panels (32 K-rows) double-buffered in LDS, filled by TENSOR_LOAD_TO_LDS
// issued from wave 0 and fenced with s_wait_tensorcnt + workgroup barrier.
// ---------------------------------------------------------------------------
__global__ __launch_bounds__(256)
void gemm_wmma_f32(const float* __restrict__ A, int lda,
                   const float* __restrict__ W,
                   float* __restrict__ Out) {
    __shared__ float smem[SMEM_FLOATS];   // 72 KB, at LDS offset 0

    const int tid  = threadIdx.x;
    const int lane = tid & 31;
    const int wv   = tid >> 5;       // wave id 0..7
    const int rowg = wv >> 2;        // 0..1
    const int colg = wv & 3;         // 0..3
    const int m    = lane & 15;
    const int hi   = lane >> 4;
    const int blockRow = blockIdx.x * 32;
    const int colBase  = colg * 64;

    v8f acc[4];
#pragma unroll
    for (int t = 0; t < 4; ++t)
#pragma unroll
        for (int v = 0; v < 8; ++v) acc[t][v] = 0.0f;

#if __has_builtin(__builtin_amdgcn_tensor_load_to_lds)
    const bool issuer = (tid == 0);
    if (issuer) {   // prologue: DMA panel 0 into buffer 0
        tdm_load_2d(W, (unsigned)(WBUF_F(0) * 4), DD, KPANEL, DD);
        tdm_load_2d(A + (size_t)blockRow * lda, (unsigned)(ABUF_F(0) * 4),
                    KPANEL, 32, lda);
    }
    for (int pi = 0; pi < NPANEL; ++pi) {
        const int buf = pi & 1;
        if (pi + 1 < NPANEL) {
            if (issuer) {
                const int pn = (pi + 1) * KPANEL;
                tdm_load_2d(W + (size_t)pn * DD,
                            (unsigned)(WBUF_F(buf ^ 1) * 4), DD, KPANEL, DD);
                tdm_load_2d(A + (size_t)blockRow * lda + pn,
                            (unsigned)(ABUF_F(buf ^ 1) * 4), KPANEL, 32, lda);
                __builtin_amdgcn_s_wait_tensorcnt(2);  // panel pi has landed
            }
        } else {
            if (issuer) __builtin_amdgcn_s_wait_tensorcnt(0);
        }
        __syncthreads();      // publish panel pi to all waves
        lds_publish(smem);    // TDM wrote smem: make the compiler believe it

        const float* wp = &smem[WBUF_F(buf)];
        const float* ap = &smem[ABUF_F(buf)];
#pragma unroll
        for (int kk = 0; kk < KPANEL; kk += 4) {
            const int kr = kk + 2 * hi;
            v2f a;   // A frag 16x4: lane m = row M; K = 2*hi + {0,1}
            a.x = ap[(rowg * 16 + m) * KPANEL + kr];
            a.y = ap[(rowg * 16 + m) * KPANEL + kr + 1];
#pragma unroll
            for (int t = 0; t < 4; ++t) {
                const int c = colBase + t * 16 + m;
                v2f b;   // B frag 4x16: lane m = col N; K = 2*hi + {0,1}
                b.x = wp[kr * DD + c];
                b.y = wp[(kr + 1) * DD + c];
                acc[t] = __builtin_amdgcn_wmma_f32_16x16x4_f32(
                    false, a, false, b, (short)0, acc[t], false, false);
            }
        }
        __syncthreads();   // all waves done with buf before it is re-filled
    }
#else
    // Fallback: manual LDS staging (no TDM builtin available)
    for (int pi = 0; pi < NPANEL; ++pi) {
        const int p = pi * KPANEL;
        {
            const float4* src = (const float4*)(W + (size_t)p * DD);
            float4* dst = (float4*)&smem[WBUF_F(0)];
            for (int i = tid; i < (KPANEL * DD) / 4; i += 256) dst[i] = src[i];
        }
        {
            const int r = tid >> 3;
            const int c = (tid & 7) * 4;
            const float* srow = A + (size_t)(blockRow + r) * lda + p + c;
            *(float4*)(&smem[ABUF_F(0) + r * KPANEL + c]) = *(const float4*)srow;
        }
        __syncthreads();
        const float* wp = &smem[WBUF_F(0)];
        const float* ap = &smem[ABUF_F(0)];
#pragma unroll
        for (int kk = 0; kk < KPANEL; kk += 4) {
            const int kr = kk + 2 * hi;
            v2f a;
            a.x = ap[(rowg * 16 + m) * KPANEL + kr];
            a.y = ap[(rowg * 16 + m) * KPANEL + kr + 1];
#pragma unroll
            for (int t = 0; t < 4; ++t) {
                const int c = colBase + t * 16 + m;
                v2f b;
                b.x = wp[kr * DD + c];
                b.y = wp[(kr + 1) * DD + c];
                acc[t] = __builtin_amdgcn_wmma_f32_16x16x4_f32(
                    false, a, false, b, (short)0, acc[t], false, false);
            }
        }
        __syncthreads();
    }
#endif

    // D layout: VGPR v holds (M = v + 8*hi, N = m)
#pragma unroll
    for (int t = 0; t < 4; ++t) {
#pragma unroll
        for (int v = 0; v < 8; ++v) {
            const int row = blockRow + rowg * 16 + v + 8 * hi;
            const int col = colBase + t * 16 + m;
            Out[(size_t)row * DD + col] = acc[t][v];
        }
    }
}

// ---------------------------------------------------------------------------
// Graph structure kernels (degree / rsqrt) — computed once, reused both layers
// ---------------------------------------------------------------------------
__global__ void deg_init_kernel(float* __restrict__ deg) {
    int i = blockIdx.x * blockDim.x + threadIdx.x;
    if (i < NT) deg[i] = 1.0f;   // +1.0 from the reference
}

__global__ void deg_scatter_kernel(const int* __restrict__ knn,
                                   const unsigned char* __restrict__ mask,
                                   float* __restrict__ deg) {
    int e = blockIdx.x * blockDim.x + threadIdx.x;
    if (e >= NEDGE) return;
    int node = e >> 4;
    int k    = (e & 15) + 1;           // skip column 0
    int idx  = node * K1 + k;
    if (!mask[idx]) return;
    int b    = node >> 12;             // N = 4096
    int dst  = knn[idx] + (b << 12);
    unsafeAtomicAdd(&deg[dst], 1.0f);
}

__global__ void dinv_kernel(const float* __restrict__ deg, float* __restrict__ dinv) {
    int i = blockIdx.x * blockDim.x + threadIdx.x;
    if (i < NT) dinv[i] = rsqrtf(deg[i]);
}

// ---------------------------------------------------------------------------
// agg = bias + dinv^2 * hW   (also zero the layernorm stats for this layer)
// ---------------------------------------------------------------------------
__global__ __launch_bounds__(256)
void init_agg_kernel(const float* __restrict__ hW,
                     const float* __restrict__ dinv,
                     const float* __restrict__ bias,
                     float* __restrict__ agg,
                     float* __restrict__ stats) {
    if (blockIdx.x == 0 && threadIdx.x < 2) stats[threadIdx.x] = 0.0f;
    size_t i = (size_t)blockIdx.x * blockDim.x + threadIdx.x;
    if (i >= NELEM) return;
    int node = (int)(i >> 8);
    int f    = (int)(i & 255);
    float di = dinv[node];
    agg[i] = bias[f] + di * di * hW[i];
}

// ---------------------------------------------------------------------------
// Edge scatter: agg[dst] += dinv[src]*dinv[dst] * hW[src]   (w==1 edges only)
// One block per edge, 256 threads = feature lanes. f32 atomics at L2.
// ---------------------------------------------------------------------------
__global__ __launch_bounds__(256)
void scatter_agg_kernel(const int* __restrict__ knn,
                        const unsigned char* __restrict__ mask,
                        const float* __restrict__ dinv,
                        const float* __restrict__ hW,
                        float* __restrict__ agg) {
    int e    = blockIdx.x;
    int node = e >> 4;
    int k    = (e & 15) + 1;
    int idx  = node * K1 + k;
    if (!mask[idx]) return;
    int b    = node >> 12;
    int dst  = knn[idx] + (b << 12);
    float coef = dinv[node] * dinv[dst];
    int f = threadIdx.x;
    unsafeAtomicAdd(&agg[(size_t)dst * DD + f], coef * hW[(size_t)node * DD + f]);
}

// ---------------------------------------------------------------------------
// Global layernorm stats: stats[0]=sum(x), stats[1]=sum(x^2) over all NT*256
// ---------------------------------------------------------------------------
__global__ __launch_bounds__(256)
void reduce_stats_kernel(const float* __restrict__ x, float* __restrict__ stats) {
    float s = 0.0f, sq = 0.0f;
    for (size_t i = (size_t)blockIdx.x * blockDim.x + threadIdx.x; i < NELEM;
         i += (size_t)gridDim.x * blockDim.x) {
        float v = x[i];
        s += v;
        sq += v * v;
    }
    __shared__ float ls[256], lq[256];
    ls[threadIdx.x] = s;
    lq[threadIdx.x] = sq;
    __syncthreads();
    for (int o = 128; o > 0; o >>= 1) {
        if (threadIdx.x < o) {
            ls[threadIdx.x] += ls[threadIdx.x + o];
            lq[threadIdx.x] += lq[threadIdx.x + o];
        }
        __syncthreads();
    }
    if (threadIdx.x == 0) {
        unsafeAtomicAdd(&stats[0], ls[0]);
        unsafeAtomicAdd(&stats[1], lq[0]);
    }
}

// ---------------------------------------------------------------------------
// y = prelu( g[f]*(x - mean)/(std+eps) + be[f] )   (mean/std from stats)
// ---------------------------------------------------------------------------
__global__ __launch_bounds__(256)
void norm_prelu_kernel(const float* __restrict__ x,
                       const float* __restrict__ g,
                       const float* __restrict__ be,
                       const float* __restrict__ alpha,
                       const float* __restrict__ stats,
                       float* __restrict__ y) {
    size_t i = (size_t)blockIdx.x * blockDim.x + threadIdx.x;
    if (i >= NELEM) return;
    const float invN = 1.0f / (float)NELEM;
    float mean = stats[0] * invN;
    float var  = stats[1] * invN - mean * mean;
    float stdv = sqrtf(fmaxf(var, 0.0f));
    float inv  = 1.0f / (stdv + EPS_LN);
    int f = (int)(i & 255);
    float v = g[f] * ((x[i] - mean) * inv) + be[f];
    float a = alpha[0];
    y[i] = (v >= 0.0f) ? v : a * v;
}

// ---------------------------------------------------------------------------
// Launch
// ---------------------------------------------------------------------------
extern "C" void kernel_launch(void* const* d_in, const int* in_sizes, int n_in,
                              void* d_out, int out_size, void* d_ws, size_t ws_size,
                              hipStream_t stream) {
    const float*         x    = (const float*)d_in[0];         // (NT, 2, 256)
    const int*           knn  = (const int*)d_in[1];           // (B, N, 17)
    const unsigned char* mask = (const unsigned char*)d_in[2]; // bool (B, N, 17)
    const float* W0 = (const float*)d_in[3];
    const float* b0 = (const float*)d_in[4];
    const float* g0 = (const float*)d_in[5];
    const float* be0 = (const float*)d_in[6];
    const float* a0 = (const float*)d_in[7];
    const float* W1 = (const float*)d_in[8];
    const float* b1 = (const float*)d_in[9];
    const float* g1 = (const float*)d_in[10];
    const float* be1 = (const float*)d_in[11];
    const float* a1 = (const float*)d_in[12];

    // Workspace layout: [tmp hW: NT*256 f32 = 32MB][deg: NT][dinv: NT][stats: 2]
    float* tmp   = (float*)d_ws;
    float* deg   = tmp + NELEM;
    float* dinv  = deg + NT;
    float* stats = dinv + NT;
    float* agg   = (float*)d_out;   // reused as h buffer for both layers

    const int TPB = 256;

    // Graph degree + rsqrt (shared by both layers)
    deg_init_kernel<<<NT / TPB, TPB, 0, stream>>>(deg);
    deg_scatter_kernel<<<NEDGE / TPB, TPB, 0, stream>>>(knn, mask, deg);
    dinv_kernel<<<NT / TPB, TPB, 0, stream>>>(deg, dinv);

    const int gemmBlocks = NT / 32;            // 1024
    const int elemBlocks = (int)(NELEM / TPB); // 32768

    // ---- Layer 0: input is x[:,0,:] (row stride 512 floats) ----
    gemm_wmma_f32<<<gemmBlocks, TPB, 0, stream>>>(x, 2 * DD, W0, tmp);
    init_agg_kernel<<<elemBlocks, TPB, 0, stream>>>(tmp, dinv, b0, agg, stats);
    scatter_agg_kernel<<<NEDGE, TPB, 0, stream>>>(knn, mask, dinv, tmp, agg);
    reduce_stats_kernel<<<1024, TPB, 0, stream>>>(agg, stats);
    norm_prelu_kernel<<<elemBlocks, TPB, 0, stream>>>(agg, g0, be0, a0, stats, agg);

    // ---- Layer 1: input is agg (row stride 256 floats) ----
    gemm_wmma_f32<<<gemmBlocks, TPB, 0, stream>>>(agg, DD, W1, tmp);
    init_agg_kernel<<<elemBlocks, TPB, 0, stream>>>(tmp, dinv, b1, agg, stats);
    scatter_agg_kernel<<<NEDGE, TPB, 0, stream>>>(knn, mask, dinv, tmp, agg);
    reduce_stats_kernel<<<1024, TPB, 0, stream>>>(agg, stats);
    norm_prelu_kernel<<<elemBlocks, TPB, 0, stream>>>(agg, g1, be1, a1, stats, agg);
}